// GNN_4501125726939
// MI455X (gfx1250) — compile-verified
//
#include <hip/hip_runtime.h>

// Problem constants (fixed by the reference).
#define NN    50000
#define EE    800000
#define DIN_  64
#define EIN_  16
#define HH    300
#define RR    1024
#define GGRP  256
#define DEPTH_ 5

typedef float v2f __attribute__((ext_vector_type(2)));
typedef float v8f __attribute__((ext_vector_type(8)));

// ---------------------------------------------------------------------------
// fp32 WMMA GEMM: C[M,NC] = act(A[M,K] @ B[K,NC] + bias[NC])
// act: 0 = none, 1 = relu, 2 = prelu(slope_ptr[0])
// K, NC compile-time so B loads use immediate offsets and stay in the
// global address space (global_load_b32, not flat_load).
// One wave computes a 32x64 tile: 2 M-subtiles x 4 N-subtiles, B fragments
// reused across both M-subtiles. K must be a multiple of 4 (64, 300 here).
// OOB rows/cols: addresses clamped for fault safety only — D[m][n] depends
// only on A row m / B col n, and OOB rows/cols are never stored.
// EXEC stays all-1s through v_wmma (ISA 7.12 requirement).
// ---------------------------------------------------------------------------
template <int K, int NC>
__global__ __launch_bounds__(256) void wmma_gemm_kernel(
    const float* __restrict__ A, const float* __restrict__ B,
    const float* __restrict__ bias, float* __restrict__ C,
    int M, int act, const float* __restrict__ slope_ptr)
{
  constexpr int NTILE = (NC + 63) / 64;      // 64-wide N supertiles
  const int lane = threadIdx.x & 31;
  const int wib  = threadIdx.x >> 5;         // 8 waves / block
  const int wid  = blockIdx.x * 8 + wib;
  const int mt   = wid / NTILE;
  const int nt   = wid - mt * NTILE;
  const int m0   = mt << 5;                  // 32 rows / wave
  if (m0 >= M) return;                       // uniform per wave
  const int n0   = nt << 6;

  const int half = lane >> 4;                // 0: K=0,1 ; 1: K=2,3
  const int l16  = lane & 15;

  // A fragment base pointers (row-major, lda = K), row-clamped
  const int r0 = m0 + l16;
  const int r1 = r0 + 16;
  const float* pA0 = A + (size_t)((r0 < M) ? r0 : (M - 1)) * K + (half << 1);
  const float* pA1 = A + (size_t)((r1 < M) ? r1 : (M - 1)) * K + (half << 1);

  // B fragment base pointers (row-major, ldb = NC), col-clamped.
  // Kept as distinct scalar pointers so addrspace inference -> global loads.
  const int c0 = n0 + l16;
  const int c1 = c0 + 16;
  const int c2 = c0 + 32;
  const int c3 = c0 + 48;
  const size_t brow = (size_t)(half << 1) * NC;
  const float* pB0 = B + brow + ((c0 < NC) ? c0 : (NC - 1));
  const float* pB1 = B + brow + ((c1 < NC) ? c1 : (NC - 1));
  const float* pB2 = B + brow + ((c2 < NC) ? c2 : (NC - 1));
  const float* pB3 = B + brow + ((c3 < NC) ? c3 : (NC - 1));

  v8f acc0[4] = {};
  v8f acc1[4] = {};

#pragma unroll 4
  for (int k0 = 0; k0 < K; k0 += 4) {
    v2f a0, a1;
    a0.x = pA0[0]; a0.y = pA0[1]; pA0 += 4;
    a1.x = pA1[0]; a1.y = pA1[1]; pA1 += 4;

    v2f b0; b0.x = pB0[0]; b0.y = pB0[NC]; pB0 += 4 * NC;
    acc0[0] = __builtin_amdgcn_wmma_f32_16x16x4_f32(false, a0, false, b0, (short)0, acc0[0], false, false);
    acc1[0] = __builtin_amdgcn_wmma_f32_16x16x4_f32(false, a1, false, b0, (short)0, acc1[0], false, false);

    v2f b1; b1.x = pB1[0]; b1.y = pB1[NC]; pB1 += 4 * NC;
    acc0[1] = __builtin_amdgcn_wmma_f32_16x16x4_f32(false, a0, false, b1, (short)0, acc0[1], false, false);
    acc1[1] = __builtin_amdgcn_wmma_f32_16x16x4_f32(false, a1, false, b1, (short)0, acc1[1], false, false);

    v2f b2; b2.x = pB2[0]; b2.y = pB2[NC]; pB2 += 4 * NC;
    acc0[2] = __builtin_amdgcn_wmma_f32_16x16x4_f32(false, a0, false, b2, (short)0, acc0[2], false, false);
    acc1[2] = __builtin_amdgcn_wmma_f32_16x16x4_f32(false, a1, false, b2, (short)0, acc1[2], false, false);

    v2f b3; b3.x = pB3[0]; b3.y = pB3[NC]; pB3 += 4 * NC;
    acc0[3] = __builtin_amdgcn_wmma_f32_16x16x4_f32(false, a0, false, b3, (short)0, acc0[3], false, false);
    acc1[3] = __builtin_amdgcn_wmma_f32_16x16x4_f32(false, a1, false, b3, (short)0, acc1[3], false, false);
  }

  float slope = 0.0f;
  if (act == 2) slope = slope_ptr[0];

  const int colv[4] = {c0, c1, c2, c3};
#pragma unroll
  for (int t = 0; t < 4; ++t) {
    const int col = colv[t];
    if (col >= NC) continue;
    const float bv = bias[col];
#pragma unroll
    for (int v = 0; v < 8; ++v) {
      const int row0 = m0 + v + (half << 3);
      if (row0 < M) {
        float val = acc0[t][v] + bv;
        if (act == 1)      val = fmaxf(val, 0.0f);
        else if (act == 2) val = (val >= 0.0f) ? val : slope * val;
        C[(size_t)row0 * NC + col] = val;
      }
      const int row1 = row0 + 16;
      if (row1 < M) {
        float val = acc1[t][v] + bv;
        if (act == 1)      val = fmaxf(val, 0.0f);
        else if (act == 2) val = (val >= 0.0f) ? val : slope * val;
        C[(size_t)row1 * NC + col] = val;
      }
    }
  }
}

// ---------------------------------------------------------------------------
// Fused GINE edge kernel: for each (edge, 4-col chunk):
//   e4 = b_pe[c..c+3] + edge_attr[ed,:16] @ W_pe[:16, c..c+3]   (recomputed;
//        avoids a 960MB e tensor — edge_attr+h+agg fit in 192MB L2)
//   m  = relu(h[src] + e4);  atomicAdd(agg[dst], m)
// agg is pre-seeded with h, so after this kernel agg == z = h + sum(m).
// ---------------------------------------------------------------------------
__global__ __launch_bounds__(256) void edge_gine_kernel(
    const float* __restrict__ h, const float* __restrict__ ea,
    const int* __restrict__ src, const int* __restrict__ dst,
    const float* __restrict__ Wpe, const float* __restrict__ bpe,
    float* __restrict__ agg)
{
  const unsigned tid   = blockIdx.x * 256u + threadIdx.x;
  const unsigned total = (unsigned)EE * 75u;   // 60,000,000
  if (tid >= total) return;
  const unsigned ed = tid / 75u;
  const unsigned c4 = tid - ed * 75u;
  const int c = (int)(c4 << 2);

  const int s = src[ed];
  const int d = dst[ed];

  float4 e4 = *(const float4*)(bpe + c);
  const float* ear = ea + (size_t)ed * EIN_;
#pragma unroll
  for (int k = 0; k < EIN_; ++k) {
    const float av = ear[k];
    const float4 w = *(const float4*)(Wpe + (size_t)k * HH + c);
    e4.x = fmaf(av, w.x, e4.x);
    e4.y = fmaf(av, w.y, e4.y);
    e4.z = fmaf(av, w.z, e4.z);
    e4.w = fmaf(av, w.w, e4.w);
  }

  const float4 hv = *(const float4*)(h + (size_t)s * HH + c);
  float4 m;
  m.x = fmaxf(hv.x + e4.x, 0.0f);
  m.y = fmaxf(hv.y + e4.y, 0.0f);
  m.z = fmaxf(hv.z + e4.z, 0.0f);
  m.w = fmaxf(hv.w + e4.w, 0.0f);

  float* outp = agg + (size_t)d * HH + c;
  unsafeAtomicAdd(outp + 0, m.x);
  unsafeAtomicAdd(outp + 1, m.y);
  unsafeAtomicAdd(outp + 2, m.z);
  unsafeAtomicAdd(outp + 3, m.w);
}

__global__ __launch_bounds__(256) void zero_kernel(float* __restrict__ p, unsigned n)
{
  unsigned i = blockIdx.x * 256u + threadIdx.x;
  if (i < n) p[i] = 0.0f;
}

// dst = src  (float4), used to seed agg with h before the edge scatter
__global__ __launch_bounds__(256) void copy_kernel(
    float* __restrict__ dstp, const float* __restrict__ srcp, unsigned n4)
{
  unsigned i = blockIdx.x * 256u + threadIdx.x;
  if (i < n4) ((float4*)dstp)[i] = ((const float4*)srcp)[i];
}

// pooled[batch[n], :] += h[n, :]   (pooled is 307KB -> L2-resident atomics)
__global__ __launch_bounds__(256) void pool_kernel(
    const float* __restrict__ h, const int* __restrict__ batch,
    float* __restrict__ pooled)
{
  const unsigned tid   = blockIdx.x * 256u + threadIdx.x;
  const unsigned total = (unsigned)NN * 75u;
  if (tid >= total) return;
  const unsigned nid = tid / 75u;
  const unsigned c4  = tid - nid * 75u;
  const int c = (int)(c4 << 2);
  const int g = batch[nid];
  const float4 hv = *(const float4*)(h + (size_t)nid * HH + c);
  float* outp = pooled + (size_t)g * HH + c;
  unsafeAtomicAdd(outp + 0, hv.x);
  unsafeAtomicAdd(outp + 1, hv.y);
  unsafeAtomicAdd(outp + 2, hv.z);
  unsafeAtomicAdd(outp + 3, hv.w);
}

// ---------------------------------------------------------------------------
static inline int gemm_blocks(int M, int NC) {
  int mt = (M + 31) / 32;
  int nt = (NC + 63) / 64;
  long waves = (long)mt * nt;
  return (int)((waves + 7) / 8);
}

extern "C" void kernel_launch(void* const* d_in, const int* in_sizes, int n_in,
                              void* d_out, int out_size, void* d_ws, size_t ws_size,
                              hipStream_t stream) {
  (void)in_sizes; (void)n_in; (void)out_size; (void)ws_size;

  const float* x         = (const float*)d_in[0];
  const float* edge_attr = (const float*)d_in[1];
  const int*   edge_index= (const int*)  d_in[2];
  const int*   batch     = (const int*)  d_in[3];
  const float* W_pn      = (const float*)d_in[4];
  const float* b_pn      = (const float*)d_in[5];
  const float* W_pe      = (const float*)d_in[6];
  const float* b_pe      = (const float*)d_in[7];
  const float* W1        = (const float*)d_in[8];
  const float* b1        = (const float*)d_in[9];
  const float* W2        = (const float*)d_in[10];
  const float* b2        = (const float*)d_in[11];
  const float* W_sp      = (const float*)d_in[12];
  const float* b_sp      = (const float*)d_in[13];
  const float* prelu_a   = (const float*)d_in[14];
  float* out = (float*)d_out;

  const int* src = edge_index;        // edge_index[0, :]
  const int* dst = edge_index + EE;   // edge_index[1, :]

  // Workspace carve-out (~180.3 MB total)
  char* ws = (char*)d_ws;
  size_t off = 0;
  auto carve = [&](size_t bytes) -> float* {
    float* p = (float*)(ws + off);
    off += (bytes + 255) & ~(size_t)255;
    return p;
  };
  float* hbuf   = carve((size_t)NN * HH * 4);
  float* agg    = carve((size_t)NN * HH * 4);
  float* tbuf   = carve((size_t)NN * HH * 4);
  float* pooled = carve((size_t)GGRP * HH * 4);

  // h = relu(x @ W_pn + b_pn)
  wmma_gemm_kernel<DIN_, HH><<<gemm_blocks(NN, HH), 256, 0, stream>>>(
      x, W_pn, b_pn, hbuf, NN, 1, prelu_a);

  const unsigned eThreads = (unsigned)EE * 75u;
  const unsigned eBlocks  = (eThreads + 255u) / 256u;
  const unsigned nh4      = (unsigned)NN * HH / 4u;

  for (int i = 0; i < DEPTH_; ++i) {
    // agg := h, then scatter adds -> agg == z = h + sum_j relu(h[src]+e)
    copy_kernel<<<(nh4 + 255u) / 256u, 256, 0, stream>>>(agg, hbuf, nh4);
    edge_gine_kernel<<<eBlocks, 256, 0, stream>>>(hbuf, edge_attr, src, dst,
                                                  W_pe, b_pe, agg);
    // t = relu(z @ W1[i] + b1[i]) ; h' = t @ W2[i] + b2[i] (relu except last)
    wmma_gemm_kernel<HH, HH><<<gemm_blocks(NN, HH), 256, 0, stream>>>(
        agg, W1 + (size_t)i * HH * HH, b1 + (size_t)i * HH, tbuf, NN, 1, prelu_a);
    wmma_gemm_kernel<HH, HH><<<gemm_blocks(NN, HH), 256, 0, stream>>>(
        tbuf, W2 + (size_t)i * HH * HH, b2 + (size_t)i * HH, hbuf, NN,
        (i < DEPTH_ - 1) ? 1 : 0, prelu_a);
  }

  const unsigned pElems = (unsigned)GGRP * HH;
  zero_kernel<<<(pElems + 255u) / 256u, 256, 0, stream>>>(pooled, pElems);
  pool_kernel<<<((unsigned)NN * 75u + 255u) / 256u, 256, 0, stream>>>(hbuf, batch, pooled);

  // out = prelu(pooled @ W_sp + b_sp)
  wmma_gemm_kernel<HH, RR><<<gemm_blocks(GGRP, RR), 256, 0, stream>>>(
      pooled, W_sp, b_sp, out, GGRP, 2, prelu_a);
}